// PolicyValue_18571438588004
// MI455X (gfx1250) — compile-verified
//
#include <hip/hip_runtime.h>
#include <hip/hip_bf16.h>

typedef __attribute__((ext_vector_type(2))) float v2f;
typedef __attribute__((ext_vector_type(8))) float v8f;

#define MB 8
#define D1C 32
#define D2C 32
#define EC 256
#define LC 16384
#define HC 256
#define NSPEC 32
#define SIDXC 8
#define NPLAY 4
#define POLW (LC + SIDXC) /* 16392 */

// workspace layout (floats): [0..7] cls_dot(m)+bp ; [64 .. 64+8192*2) emb_dot[(m*1024+d1*32+d2)*2 + k]

__device__ __forceinline__ float dot8(float4 a, float4 b, float4 c, float4 d) {
    return a.x*b.x + a.y*b.y + a.z*b.z + a.w*b.w +
           c.x*d.x + c.y*d.y + c.z*d.z + c.w*d.w;
}

__device__ __forceinline__ float wave_sum(float v) {
    #pragma unroll
    for (int off = 16; off > 0; off >>= 1) v += __shfl_xor(v, off, 32);
    return v;
}

// One 16x16 f32 tile, K=256, via V_WMMA_F32_16X16X4_F32.
// A: row-major, leading dim 256, 16 rows (padded) -- LDS resident, unconditional
//    8-byte ds_load_b64 per step (kA always even).
// B: row-major [256 x ldb], column bcol -- unconditional loads (caller guarantees valid).
__device__ __forceinline__ v8f wmma_tile_k256(const float* A, const float* B,
                                              int ldb, int bcol, int arow, int khalf) {
    v8f acc = {};
    #pragma unroll 8
    for (int k0 = 0; k0 < 256; k0 += 4) {
        const int kA = k0 + khalf * 2;
        v2f a = *(const v2f*)(A + arow * 256 + kA);   // A[row][kA], A[row][kA+1]
        v2f b;
        b.x = B[(kA    ) * ldb + bcol];               // B[kA][col]
        b.y = B[(kA + 1) * ldb + bcol];               // B[kA+1][col]
        acc = __builtin_amdgcn_wmma_f32_16x16x4_f32(
                  false, a, false, b, (short)0, acc, false, false);
    }
    return acc;
}

// ---------------- Kernel A: per-position policy dot precompute + cls dot ----------------
__global__ __launch_bounds__(256) void emb_dot_kernel(
    const float* __restrict__ emb, const float* __restrict__ cls,
    const float* __restrict__ Wp, const float* __restrict__ bp,
    float* __restrict__ ws)
{
    const int lane = threadIdx.x & 31;
    const int wave = threadIdx.x >> 5;
    const int row  = blockIdx.x * 8 + wave;          // 0..8191 = m*1024 + d1*32 + d2
    float* emb_dot = ws + 64;

    const float4* e4  = (const float4*)(emb + (size_t)row * EC);
    const float4* w04 = (const float4*)(Wp);
    const float4* w14 = (const float4*)(Wp + EC);

    float4 ea = e4[lane*2 + 0], eb = e4[lane*2 + 1];
    float p0 = dot8(ea, w04[lane*2+0], eb, w04[lane*2+1]);
    float p1 = dot8(ea, w14[lane*2+0], eb, w14[lane*2+1]);
    p0 = wave_sum(p0);
    p1 = wave_sum(p1);
    if (lane == 0) { emb_dot[row*2 + 0] = p0; emb_dot[row*2 + 1] = p1; }

    // block 0 / wave 0 additionally computes cls_dot[m] = dot(cls[m], Wp[512:768]) + bp
    if (blockIdx.x == 0 && wave == 0) {
        const float4* wc4 = (const float4*)(Wp + 2*EC);
        float4 wca = wc4[lane*2], wcb = wc4[lane*2+1];
        float bias = bp[0];
        for (int m = 0; m < MB; ++m) {
            const float4* c4 = (const float4*)(cls + m*EC);
            float p = dot8(c4[lane*2], wca, c4[lane*2+1], wcb);
            p = wave_sum(p);
            if (lane == 0) ws[m] = p + bias;
        }
    }
}

// ---------------- Kernel B: MLP heads via V_WMMA_F32_16X16X4_F32 ----------------
// single workgroup, 16 waves. M=8 padded to 16 rows; all A fragments + the value-head
// B matrix staged zero-padded in LDS so every WMMA operand load is unconditional.
__global__ __launch_bounds__(512) void heads_kernel(
    const float* __restrict__ cls,
    const float* __restrict__ Wv1, const float* __restrict__ bv1,
    const float* __restrict__ Wv2, const float* __restrict__ bv2,
    const float* __restrict__ Ws1, const float* __restrict__ bs1,
    const float* __restrict__ Ws2, const float* __restrict__ bs2,
    const int*   __restrict__ sidx,
    float* __restrict__ out)
{
    __shared__ float clsLds[16 * EC];      // cls padded to 16 rows (rows 8..15 = 0)
    __shared__ float hv[16 * HC];          // relu(cls@Wv1+bv1), padded rows zero
    __shared__ float hs[16 * HC];          // relu(cls@Ws1+bs1), padded rows zero
    __shared__ float w2v[HC * 16];         // Wv2 padded 256x4 -> 256x16 (cols 4..15 = 0)
    __shared__ float spAll[MB * NSPEC];    // special_all

    const int t     = threadIdx.x;
    const int lane  = t & 31;
    const int wave  = t >> 5;              // 0..15
    const int colr  = lane & 15;           // row for A / col for B,D
    const int khalf = lane >> 4;           // K-half select per ISA f32 A/B layout

    // ---- stage + zero-pad LDS operands ----
    for (int i = t; i < 16 * EC; i += 512) {
        const int r = i >> 8, c = i & 255;
        clsLds[i] = (r < MB) ? cls[r * EC + c] : 0.f;
    }
    for (int i = t; i < 8 * HC; i += 512) {        // zero pad rows 8..15 of hiddens
        hv[8 * HC + i] = 0.f;
        hs[8 * HC + i] = 0.f;
    }
    for (int i = t; i < HC * 16; i += 512) {       // pad Wv2 to 16 cols
        const int k = i >> 4, c = i & 15;
        w2v[i] = (c < NPLAY) ? Wv2[k * NPLAY + c] : 0.f;
    }
    __syncthreads();

    // ---- layer 1: 32 tiles (16 value-hidden, 16 special-hidden) over 16 waves ----
    for (int job = wave; job < 32; job += 16) {
        const float* W1 = (job < 16) ? Wv1 : Ws1;
        const float* b1 = (job < 16) ? bv1 : bs1;
        float* dst      = (job < 16) ? hv  : hs;
        const int n0 = (job & 15) * 16;
        v8f acc = wmma_tile_k256(clsLds, W1, HC, n0 + colr, colr, khalf);
        if (khalf == 0) {                  // lanes 0-15 hold rows 0..7 in acc[i]
            const float bias = b1[n0 + colr];
            #pragma unroll
            for (int i = 0; i < MB; ++i) {
                const float v = acc[i] + bias;
                dst[i * HC + n0 + colr] = v > 0.f ? v : 0.f;
            }
        }
    }
    __syncthreads();

    // ---- layer 2: waves 0,1 -> special (two 16-col tiles); wave 2 -> value ----
    if (wave < 2) {
        const int n0 = wave * 16;          // NSPEC=32: both tiles fully valid
        v8f acc = wmma_tile_k256(hs, Ws2, NSPEC, n0 + colr, colr, khalf);
        if (khalf == 0) {
            const float bias = bs2[n0 + colr];
            #pragma unroll
            for (int i = 0; i < MB; ++i)
                spAll[i * NSPEC + n0 + colr] = acc[i] + bias;
        }
    } else if (wave == 2) {
        v8f acc = wmma_tile_k256(hv, w2v, 16, colr, colr, khalf);
        if (khalf == 0 && colr < NPLAY) {
            const float bias = bv2[colr];
            #pragma unroll
            for (int i = 0; i < MB; ++i)
                out[MB * POLW + i * NPLAY + colr] = acc[i] + bias;  // value output
        }
    }
    __syncthreads();

    // ---- scatter special_all[:, sidx] into the policy tail ----
    if (t < MB * SIDXC) {
        const int m = t >> 3, j = t & 7;
        out[m * POLW + LC + j] = spAll[m * NSPEC + sidx[j]];
    }
}

// ---------------- Kernel C: policy assembly (pure gather + 3 adds) ----------------
__global__ __launch_bounds__(256) void policy_kernel(
    const int* __restrict__ moves, const float* __restrict__ ws,
    float* __restrict__ out)
{
    __shared__ float clsd[MB];
    if (threadIdx.x < MB) clsd[threadIdx.x] = ws[threadIdx.x];
    __syncthreads();

    const int l = blockIdx.x * blockDim.x + threadIdx.x;  // 0..16383
    const float* emb_dot = ws + 64;
    const int4 mv = ((const int4*)moves)[l];  // {d1_0, d2_0, d1_1, d2_1}
    const int r0 = mv.x * D2C + mv.y;
    const int r1 = mv.z * D2C + mv.w;
    #pragma unroll
    for (int m = 0; m < MB; ++m) {
        const float d0 = emb_dot[(m*1024 + r0)*2 + 0];
        const float d1 = emb_dot[(m*1024 + r1)*2 + 1];
        out[m*POLW + l] = d0 + d1 + clsd[m];
    }
}

extern "C" void kernel_launch(void* const* d_in, const int* in_sizes, int n_in,
                              void* d_out, int out_size, void* d_ws, size_t ws_size,
                              hipStream_t stream) {
    (void)in_sizes; (void)n_in; (void)out_size; (void)ws_size;
    const float* emb  = (const float*)d_in[0];
    const float* cls  = (const float*)d_in[1];
    const int*   mov  = (const int*)  d_in[2];
    const int*   sidx = (const int*)  d_in[3];
    const float* Wp   = (const float*)d_in[4];
    const float* bp   = (const float*)d_in[5];
    const float* Wv1  = (const float*)d_in[6];
    const float* bv1  = (const float*)d_in[7];
    const float* Wv2  = (const float*)d_in[8];
    const float* bv2  = (const float*)d_in[9];
    const float* Ws1  = (const float*)d_in[10];
    const float* bs1  = (const float*)d_in[11];
    const float* Ws2  = (const float*)d_in[12];
    const float* bs2  = (const float*)d_in[13];
    float* out = (float*)d_out;
    float* ws  = (float*)d_ws;

    // A: 8192 embedding positions, 8 waves/block -> 1024 blocks (streams 8MB once)
    emb_dot_kernel<<<1024, 256, 0, stream>>>(emb, cls, Wp, bp, ws);
    // B: single workgroup, 16 waves of f32 WMMA
    heads_kernel<<<1, 512, 0, stream>>>(cls, Wv1, bv1, Wv2, bv2,
                                        Ws1, bs1, Ws2, bs2, sidx, out);
    // C: 16384 policy columns, 8 outputs each
    policy_kernel<<<LC/256, 256, 0, stream>>>(mov, ws, out);
}